// MovieHeteroGAT_57123065037487
// MI455X (gfx1250) — compile-verified
//
#include <hip/hip_runtime.h>
#include <hip/hip_bf16.h>

// ---------------------------------------------------------------------------
// MovieHeteroGAT on gfx1250 (MI455X).
// Dense projections: V_WMMA_F32_16X16X4_F32 with LDS-staged B tiles.
// Edge softmax/scatter uses atomics; LayerNorm uses wave32 shuffles.
// ---------------------------------------------------------------------------

typedef float v2f __attribute__((ext_vector_type(2)));
typedef float v8f __attribute__((ext_vector_type(8)));

#define LEAK 0.2f

// Monotone unsigned key for float (radix-sort trick) so we can use atomicMax(u32).
__device__ __forceinline__ unsigned fenc(float f) {
    unsigned u = __float_as_uint(f);
    return (u & 0x80000000u) ? ~u : (u | 0x80000000u);
}
__device__ __forceinline__ float fdec(unsigned k) {
    unsigned u = (k & 0x80000000u) ? (k ^ 0x80000000u) : ~k;
    return __uint_as_float(u);
}

// ---------------------------------------------------------------------------
// Y[M,N] = gather(X)[M,K] @ W[K,N] + bias.   gidx==nullptr -> identity gather.
// Block = 256 threads = 8 waves. Each wave computes a 16x64 tile (4 WMMA accums).
// All 8 waves share one K-chunk x 64-col W tile staged in LDS, packed so each
// 16x16 B fragment is a single ds_load_b64:
//   lw[(k>>1)*128 + col*2 + (k&1)] = W[k][n0+col]
// A layout (16x4 f32): lanes 0-15 hold M=0..15 / K=k,k+1 ; lanes 16-31 K=k+2,k+3.
// D layout (16x16 f32): VGPR r -> row r (lanes 0-15) / row r+8 (lanes 16-31).
// M % 16 == 0, N % 64 == 0, K % 4 == 0 (true for all call sites here).
// ---------------------------------------------------------------------------
#define KCHUNK 128

__global__ void __launch_bounds__(256)
wmma_gemm_gather(const float* __restrict__ X, const int* __restrict__ gidx,
                 const float* __restrict__ W, const float* __restrict__ bias,
                 float* __restrict__ Y, int M, int K, int N)
{
    __shared__ float lw[(KCHUNK / 2) * 128];   // 32 KB

    const int t    = threadIdx.x;
    const int wave = t >> 5;
    const int lane = t & 31;
    const int half = lane >> 4;      // 0 or 1
    const int l16  = lane & 15;

    const int m0 = (blockIdx.x * 8 + wave) * 16;
    const int n0 = blockIdx.y * 64;
    const bool act = (m0 < M);       // wave-uniform: whole 16-row tile valid or not

    int row  = m0 + l16;
    int srow = 0;
    if (act) srow = gidx ? gidx[row] : row;
    const float* xrow = X + (size_t)srow * K;

    v8f c0 = {}, c1 = {}, c2 = {}, c3 = {};

    for (int k0 = 0; k0 < K; k0 += KCHUNK) {
        const int KC = (K - k0 < KCHUNK) ? (K - k0) : KCHUNK;
        __syncthreads();             // previous chunk fully consumed
        // cooperative coalesced fill of the W tile
        for (int j = t; j < KC * 64; j += 256) {
            int kl  = j >> 6;
            int col = j & 63;
            lw[(kl >> 1) * 128 + col * 2 + (kl & 1)] =
                W[(size_t)(k0 + kl) * N + n0 + col];
        }
        __syncthreads();

        if (act) {
            for (int k = 0; k < KC; k += 4) {
                const int kl = k + (half << 1);          // even
                v2f a;
                a.x = xrow[k0 + kl];
                a.y = xrow[k0 + kl + 1];
                const float* lb = lw + (kl >> 1) * 128 + (l16 << 1);
                v2f b0 = *(const v2f*)(lb);
                v2f b1 = *(const v2f*)(lb + 32);
                v2f b2 = *(const v2f*)(lb + 64);
                v2f b3 = *(const v2f*)(lb + 96);
                c0 = __builtin_amdgcn_wmma_f32_16x16x4_f32(false, a, false, b0, (short)0, c0, false, false);
                c1 = __builtin_amdgcn_wmma_f32_16x16x4_f32(false, a, false, b1, (short)0, c1, false, false);
                c2 = __builtin_amdgcn_wmma_f32_16x16x4_f32(false, a, false, b2, (short)0, c2, false, false);
                c3 = __builtin_amdgcn_wmma_f32_16x16x4_f32(false, a, false, b3, (short)0, c3, false, false);
            }
        }
    }

    if (act) {
        const int rbase = m0 + (half << 3);
        #pragma unroll
        for (int r = 0; r < 8; ++r) {
            float* yr = Y + (size_t)(rbase + r) * N + n0 + l16;
            yr[0]  = c0[r] + bias[n0 + l16];
            yr[16] = c1[r] + bias[n0 + 16 + l16];
            yr[32] = c2[r] + bias[n0 + 32 + l16];
            yr[48] = c3[r] + bias[n0 + 48 + l16];
        }
    }
}

// ---------------------------------------------------------------------------
// Fill helpers (grid-stride)
// ---------------------------------------------------------------------------
__global__ void fill_f32(float* __restrict__ p, float v, long n) {
    long i  = (long)blockIdx.x * blockDim.x + threadIdx.x;
    long st = (long)gridDim.x * blockDim.x;
    for (; i < n; i += st) p[i] = v;
}
__global__ void fill_u32(unsigned* __restrict__ p, unsigned v, long n) {
    long i  = (long)blockIdx.x * blockDim.x + threadIdx.x;
    long st = (long)gridDim.x * blockDim.x;
    for (; i < n; i += st) p[i] = v;
}

// ---------------------------------------------------------------------------
// Edge stage 1: logits + segment max.
// One thread per (edge, head). HC = 256, H = 4, C = 64 for every GAT layer.
// logit = sum_c att[h][c] * leaky( xl[s][h*64+c] + xr[d][h*64+c] + ea*We[h*64+c] )
// ---------------------------------------------------------------------------
__global__ void __launch_bounds__(256)
edge_logits(const int* __restrict__ ei, const float* __restrict__ ea,
            const float* __restrict__ xl, const float* __restrict__ xr,
            const float* __restrict__ We, const float* __restrict__ att,
            float* __restrict__ logit, unsigned* __restrict__ mxkey, int E_)
{
    int idx = blockIdx.x * blockDim.x + threadIdx.x;
    if (idx >= E_ * 4) return;
    int e = idx >> 2, h = idx & 3;
    int s = ei[e], d = ei[E_ + e];
    float w = ea[e];
    const float* xls = xl + (size_t)s * 256 + h * 64;
    const float* xrd = xr + (size_t)d * 256 + h * 64;
    const float* Wh  = We + h * 64;
    const float* ah  = att + h * 64;
    float acc = 0.f;
    #pragma unroll 8
    for (int c = 0; c < 64; ++c) {
        float v = xls[c] + xrd[c] + w * Wh[c];
        v = v > 0.f ? v : LEAK * v;
        acc += ah[c] * v;
    }
    logit[idx] = acc;
    atomicMax(mxkey + (d * 4 + h), fenc(acc));
}

// ---------------------------------------------------------------------------
// Edge stage 2: ex = exp(logit - mx[d]); den[d] += ex
// ---------------------------------------------------------------------------
__global__ void __launch_bounds__(256)
edge_expsum(const int* __restrict__ ei, const float* __restrict__ logit,
            const unsigned* __restrict__ mxkey, float* __restrict__ ex,
            float* __restrict__ den, int E_)
{
    int idx = blockIdx.x * blockDim.x + threadIdx.x;
    if (idx >= E_ * 4) return;
    int e = idx >> 2, h = idx & 3;
    int d = ei[E_ + e];
    float m = fdec(mxkey[d * 4 + h]);
    float v = __expf(logit[idx] - m);
    ex[idx] = v;
    atomicAdd(den + d * 4 + h, v);
}

// ---------------------------------------------------------------------------
// Edge stage 3: out[d] += alpha * xl[s].  One block (256 thr) per edge.
// ---------------------------------------------------------------------------
__global__ void __launch_bounds__(256)
edge_scatter(const int* __restrict__ ei, const float* __restrict__ xl,
             const float* __restrict__ ex, const float* __restrict__ den,
             float* __restrict__ out, int E_)
{
    __shared__ int sd[2];
    int e = blockIdx.x;
    int c = threadIdx.x;          // 0..255
    if (c == 0) { sd[0] = ei[e]; sd[1] = ei[E_ + e]; }
    __syncthreads();
    int s = sd[0], d = sd[1];
    int h = c >> 6;
    float alpha = ex[e * 4 + h] / (den[d * 4 + h] + 1e-16f);
    atomicAdd(out + (size_t)d * 256 + c, alpha * xl[(size_t)s * 256 + c]);
}

// ---------------------------------------------------------------------------
// LayerNorm(+gat bias)+leaky over 256 channels, in place. One block per node.
// ---------------------------------------------------------------------------
__global__ void __launch_bounds__(256)
ln_act_concat(float* __restrict__ y, const float* __restrict__ bias,
              const float* __restrict__ g, const float* __restrict__ be, int n)
{
    __shared__ float red[8];
    int node = blockIdx.x;
    int t = threadIdx.x;
    float v = y[(size_t)node * 256 + t] + bias[t];
    float s = v;
    for (int o = 16; o > 0; o >>= 1) s += __shfl_down(s, o);
    if ((t & 31) == 0) red[t >> 5] = s;
    __syncthreads();
    float tot = 0.f;
    #pragma unroll
    for (int i = 0; i < 8; ++i) tot += red[i];
    float mean = tot * (1.f / 256.f);
    __syncthreads();
    float dd = v - mean;
    float s2 = dd * dd;
    for (int o = 16; o > 0; o >>= 1) s2 += __shfl_down(s2, o);
    if ((t & 31) == 0) red[t >> 5] = s2;
    __syncthreads();
    float tot2 = 0.f;
    #pragma unroll
    for (int i = 0; i < 8; ++i) tot2 += red[i];
    float var = tot2 * (1.f / 256.f);
    float o_ = g[t] * dd * rsqrtf(var + 1e-5f) + be[t];
    y[(size_t)node * 256 + t] = o_ > 0.f ? o_ : LEAK * o_;
}

// ---------------------------------------------------------------------------
// mean over 4 heads -> +bias -> LayerNorm(64) -> leaky -> out. Block per node.
// ---------------------------------------------------------------------------
__global__ void __launch_bounds__(64)
ln_act_mean(const float* __restrict__ agg, const float* __restrict__ bias,
            const float* __restrict__ g, const float* __restrict__ be,
            float* __restrict__ out, int n)
{
    __shared__ float red[2];
    int node = blockIdx.x;
    int t = threadIdx.x;          // 0..63
    const float* a = agg + (size_t)node * 256;
    float v = 0.25f * (a[t] + a[64 + t] + a[128 + t] + a[192 + t]) + bias[t];
    float s = v;
    for (int o = 16; o > 0; o >>= 1) s += __shfl_down(s, o);
    if ((t & 31) == 0) red[t >> 5] = s;
    __syncthreads();
    float mean = (red[0] + red[1]) * (1.f / 64.f);
    __syncthreads();
    float dd = v - mean;
    float s2 = dd * dd;
    for (int o = 16; o > 0; o >>= 1) s2 += __shfl_down(s2, o);
    if ((t & 31) == 0) red[t >> 5] = s2;
    __syncthreads();
    float var = (red[0] + red[1]) * (1.f / 64.f);
    float o_ = g[t] * dd * rsqrtf(var + 1e-5f) + be[t];
    out[(size_t)node * 64 + t] = o_ > 0.f ? o_ : LEAK * o_;
}

// ---------------------------------------------------------------------------
// Host side
// ---------------------------------------------------------------------------
static inline void launch_gemm(const float* X, const int* gidx, const float* W,
                               const float* bias, float* Y, int M, int K, int N,
                               hipStream_t s)
{
    int mtiles = M / 16;
    dim3 grid((mtiles + 7) / 8, N / 64);
    wmma_gemm_gather<<<grid, 256, 0, s>>>(X, gidx, W, bias, Y, M, K, N);
}

static inline void launch_fill_f(float* p, float v, long n, hipStream_t s) {
    int blocks = (int)((n + 255) / 256); if (blocks > 4096) blocks = 4096;
    fill_f32<<<blocks, 256, 0, s>>>(p, v, n);
}
static inline void launch_fill_u(unsigned* p, unsigned v, long n, hipStream_t s) {
    int blocks = (int)((n + 255) / 256); if (blocks > 4096) blocks = 4096;
    fill_u32<<<blocks, 256, 0, s>>>(p, v, n);
}

static inline void run_gat_edges(const int* ei, const float* ea,
                                 const float* xl, const float* xr,
                                 const float* We, const float* att,
                                 float* agg, int ndst, int E_,
                                 float* logit, float* ex, unsigned* mxkey, float* den,
                                 hipStream_t s)
{
    launch_fill_u(mxkey, 0u, (long)ndst * 4, s);              // key 0 == most-negative
    launch_fill_f(den, 0.f, (long)ndst * 4, s);
    launch_fill_f(agg, 0.f, (long)ndst * 256, s);
    int nth = E_ * 4;
    edge_logits<<<(nth + 255) / 256, 256, 0, s>>>(ei, ea, xl, xr, We, att, logit, mxkey, E_);
    edge_expsum<<<(nth + 255) / 256, 256, 0, s>>>(ei, logit, mxkey, ex, den, E_);
    edge_scatter<<<E_, 256, 0, s>>>(ei, xl, ex, den, agg, E_);
}

extern "C" void kernel_launch(void* const* d_in, const int* in_sizes, int n_in,
                              void* d_out, int out_size, void* d_ws, size_t ws_size,
                              hipStream_t stream)
{
    const int* user_ids  = (const int*)d_in[0];
    const int* movie_ids = (const int*)d_in[1];
    const int* ei_um     = (const int*)d_in[2];
    const int* ei_mu     = (const int*)d_in[3];
    const float* ea_um   = (const float*)d_in[4];
    const float* ea_mu   = (const float*)d_in[5];
    const float* user_emb  = (const float*)d_in[6];
    const float* movie_emb = (const float*)d_in[7];
    const float* W_user  = (const float*)d_in[8];
    const float* b_user  = (const float*)d_in[9];
    const float* W_movie = (const float*)d_in[10];
    const float* b_movie = (const float*)d_in[11];
    // GAT parameter blocks: Wl, bl, Wr, br, We, att, b
    const float* const* P = (const float* const*)d_in;
    const int L0UM = 12, L0MU = 19, L1UM = 26, L1MU = 33;
    const float* g0u  = (const float*)d_in[40];
    const float* be0u = (const float*)d_in[41];
    const float* g0m  = (const float*)d_in[42];
    const float* be0m = (const float*)d_in[43];
    const float* g1u  = (const float*)d_in[44];
    const float* be1u = (const float*)d_in[45];
    const float* g1m  = (const float*)d_in[46];
    const float* be1m = (const float*)d_in[47];

    const int NU = in_sizes[0];     // 100000
    const int NM = in_sizes[1];     // 20000
    const int E_ = in_sizes[4];     // 300000

    // --- workspace arena (fixed offsets, reused across stages) ---
    float* base = (float*)d_ws;
    size_t off = 0;
    auto alloc = [&](size_t nfl) { float* p = base + off; off += (nfl + 63) & ~(size_t)63; return p; };
    float* A    = alloc((size_t)NU * 64);    // xu               ; later xl1mu
    float* B    = alloc((size_t)NM * 64);    // xm
    float* C    = alloc((size_t)NU * 256);   // xl0um            ; later xr1mu
    float* D    = alloc((size_t)NM * 256);   // xr0um            ; later xr1um
    float* Ebuf = alloc((size_t)NM * 256);   // agg0m -> xm1
    float* F    = alloc((size_t)NM * 256);   // xl0mu            ; later agg1m
    float* G    = alloc((size_t)NU * 256);   // xr0mu            ; later xl1um ; later agg1u
    float* H    = alloc((size_t)NU * 256);   // agg0u -> xu1
    float* LOGIT = alloc((size_t)E_ * 4);
    float* EX    = alloc((size_t)E_ * 4);
    unsigned* MX = (unsigned*)alloc((size_t)NU * 4);
    float* DEN   = alloc((size_t)NU * 4);
    if (off * sizeof(float) > ws_size) return;   // constant across calls

    float* xu2 = (float*)d_out;
    float* xm2 = xu2 + (size_t)NU * 64;

    // --- input projections (gather GEMM) ---
    launch_gemm(user_emb,  user_ids,  W_user,  b_user,  A, NU, 64, 64, stream);
    launch_gemm(movie_emb, movie_ids, W_movie, b_movie, B, NM, 64, 64, stream);

    // --- layer 0, user -> movie ---
    launch_gemm(A, nullptr, P[L0UM + 0], P[L0UM + 1], C, NU, 64, 256, stream);  // xl0um
    launch_gemm(B, nullptr, P[L0UM + 2], P[L0UM + 3], D, NM, 64, 256, stream);  // xr0um
    run_gat_edges(ei_um, ea_um, C, D, P[L0UM + 4], P[L0UM + 5], Ebuf, NM, E_,
                  LOGIT, EX, MX, DEN, stream);

    // --- layer 0, movie -> user ---
    launch_gemm(B, nullptr, P[L0MU + 0], P[L0MU + 1], F, NM, 64, 256, stream);  // xl0mu
    launch_gemm(A, nullptr, P[L0MU + 2], P[L0MU + 3], G, NU, 64, 256, stream);  // xr0mu
    run_gat_edges(ei_mu, ea_mu, F, G, P[L0MU + 4], P[L0MU + 5], H, NU, E_,
                  LOGIT, EX, MX, DEN, stream);

    // --- LN + act (in place): xm1 = Ebuf, xu1 = H ---
    ln_act_concat<<<NM, 256, 0, stream>>>(Ebuf, P[L0UM + 6], g0m, be0m, NM);
    ln_act_concat<<<NU, 256, 0, stream>>>(H,    P[L0MU + 6], g0u, be0u, NU);

    // --- layer 1, user -> movie ---
    launch_gemm(H,    nullptr, P[L1UM + 0], P[L1UM + 1], G, NU, 256, 256, stream); // xl1um
    launch_gemm(Ebuf, nullptr, P[L1UM + 2], P[L1UM + 3], D, NM, 256, 256, stream); // xr1um
    run_gat_edges(ei_um, ea_um, G, D, P[L1UM + 4], P[L1UM + 5], F, NM, E_,
                  LOGIT, EX, MX, DEN, stream);
    ln_act_mean<<<NM, 64, 0, stream>>>(F, P[L1UM + 6], g1m, be1m, xm2, NM);

    // --- layer 1, movie -> user ---
    launch_gemm(Ebuf, nullptr, P[L1MU + 0], P[L1MU + 1], A, NM, 256, 256, stream); // xl1mu
    launch_gemm(H,    nullptr, P[L1MU + 2], P[L1MU + 3], C, NU, 256, 256, stream); // xr1mu
    run_gat_edges(ei_mu, ea_mu, A, C, P[L1MU + 4], P[L1MU + 5], G, NU, E_,
                  LOGIT, EX, MX, DEN, stream);
    ln_act_mean<<<NU, 64, 0, stream>>>(G, P[L1MU + 6], g1u, be1u, xu2, NU);
}